// Dynamic_Relu_C_62242666054407
// MI455X (gfx1250) — compile-verified
//
#include <hip/hip_runtime.h>
#include <math.h>

#define NSAMP 32
#define CCH   64
#define HWN   12544          // 112*112
#define HW4   3136           // HWN/4
#define HIDDEN 8
#define TWOKC 256
#define T_INV (1.0f/10.0f)
#define GAMMA (12544.0f/3.0f)

typedef __attribute__((ext_vector_type(2))) float v2f;
typedef __attribute__((ext_vector_type(8))) float v8f;

// ---------------- zero pooled accumulator ----------------
__global__ void k_zero(float* __restrict__ pooled) {
    int i = blockIdx.x * blockDim.x + threadIdx.x;
    if (i < NSAMP * CCH) pooled[i] = 0.0f;
}

// ---------------- pass 1: s[n,hw] = <x, conv_w> + b ; pooled[n,c] = sum_hw x ----------------
__global__ __launch_bounds__(256) void k_pass1(const float* __restrict__ x,
                                               const float* __restrict__ conv_w,
                                               const float* __restrict__ conv_b,
                                               float* __restrict__ s_out,
                                               float* __restrict__ pooled) {
    const int split = blockIdx.x;          // 0..7
    const int n     = blockIdx.y;          // 0..31
    const int tid   = threadIdx.x;         // 0..255
    const int lane  = tid & 31;
    const int P     = HW4 / 8;             // 392 float4 positions per split
    const int p1    = split * P + tid;
    const int p2    = split * P + 256 + tid;
    const bool ok2  = (tid < P - 256);     // tid < 136

    const float4* xn = (const float4*)x + (size_t)n * CCH * HW4;
    float4 sA = {0.f, 0.f, 0.f, 0.f};
    float4 sB = {0.f, 0.f, 0.f, 0.f};

    for (int c = 0; c < CCH; ++c) {
        const float wc = conv_w[c];
        const float4* xc = xn + (size_t)c * HW4;
        float4 v1 = xc[p1];
        float4 v2 = {0.f, 0.f, 0.f, 0.f};
        if (ok2) v2 = xc[p2];
        sA.x += v1.x * wc; sA.y += v1.y * wc; sA.z += v1.z * wc; sA.w += v1.w * wc;
        sB.x += v2.x * wc; sB.y += v2.y * wc; sB.z += v2.z * wc; sB.w += v2.w * wc;
        float cs = (v1.x + v1.y) + (v1.z + v1.w) + (v2.x + v2.y) + (v2.z + v2.w);
        #pragma unroll
        for (int off = 16; off >= 1; off >>= 1) cs += __shfl_xor(cs, off, 32);
        if (lane == 0) atomicAdd(&pooled[n * CCH + c], cs);
    }

    const float cb = conv_b[0];
    float4* so = (float4*)s_out + (size_t)n * HW4;
    sA.x += cb; sA.y += cb; sA.z += cb; sA.w += cb;
    so[p1] = sA;
    if (ok2) {
        sB.x += cb; sB.y += cb; sB.z += cb; sB.w += cb;
        so[p2] = sB;
    }
}

// ---------------- softmax over HW per sample; in-place s -> spatial = min(gamma*p, 1) ----------------
__global__ __launch_bounds__(1024) void k_softmax(float* __restrict__ s) {
    const int n   = blockIdx.x;
    const int tid = threadIdx.x;
    const int lane = tid & 31;
    const int wid  = tid >> 5;             // 0..31
    float* sn = s + (size_t)n * HWN;

    float vals[13];
    float lmax = -3.4e38f;
    #pragma unroll
    for (int i = 0; i < 13; ++i) {
        int idx = tid + i * 1024;
        float v = (idx < HWN) ? sn[idx] : -3.4e38f;
        vals[i] = v;
        lmax = fmaxf(lmax, v);
    }
    #pragma unroll
    for (int off = 16; off >= 1; off >>= 1) lmax = fmaxf(lmax, __shfl_xor(lmax, off, 32));

    __shared__ float red[32];
    __shared__ float bc[2];
    if (lane == 0) red[wid] = lmax;
    __syncthreads();
    if (wid == 0) {
        float m = red[lane];
        #pragma unroll
        for (int off = 16; off >= 1; off >>= 1) m = fmaxf(m, __shfl_xor(m, off, 32));
        if (lane == 0) bc[0] = m;
    }
    __syncthreads();
    const float m = bc[0];

    float lsum = 0.0f;
    #pragma unroll
    for (int i = 0; i < 13; ++i) {
        int idx = tid + i * 1024;
        if (idx < HWN) lsum += __expf((vals[i] - m) * T_INV);
    }
    #pragma unroll
    for (int off = 16; off >= 1; off >>= 1) lsum += __shfl_xor(lsum, off, 32);
    if (lane == 0) red[wid] = lsum;
    __syncthreads();
    if (wid == 0) {
        float ssum = red[lane];
        #pragma unroll
        for (int off = 16; off >= 1; off >>= 1) ssum += __shfl_xor(ssum, off, 32);
        if (lane == 0) bc[1] = ssum;
    }
    __syncthreads();
    const float inv = 1.0f / bc[1];

    #pragma unroll
    for (int i = 0; i < 13; ++i) {
        int idx = tid + i * 1024;
        if (idx < HWN) {
            float e = __expf((vals[i] - m) * T_INV);
            sn[idx] = fminf(GAMMA * e * inv, 1.0f);
        }
    }
}

// ---------------- coefficient MLP via WMMA f32 16x16x4 ----------------
// GEMM1: h[32,8] = relu(pooled_mean[32,64] @ w1[64,8] + b1)
// GEMM2: t[32,256] = 2*sigmoid(h @ w2[8,256] + b2) - 1  -> coef[n][c][4] = (a1,b1,a2,b2)
__global__ __launch_bounds__(512) void k_coef(const float* __restrict__ pooled,
                                              const float* __restrict__ w1,
                                              const float* __restrict__ b1v,
                                              const float* __restrict__ w2,
                                              const float* __restrict__ b2v,
                                              float* __restrict__ coef) {
    __shared__ float hbuf[32 * HIDDEN];
    const int tid   = threadIdx.x;
    const int lane  = tid & 31;
    const int wid   = tid >> 5;            // 0..15
    const int row   = lane & 15;
    const int khalf = (lane >> 4) * 2;     // K offset for this lane half (A layout)
    const int half8 = (lane >> 4) * 8;     // M offset for this lane half (D layout)

    if (wid == 0) {
        // GEMM1: M=32 (2 tiles), N=8 (in 16-wide tile), K=64 (16 steps of 4)
        for (int mt = 0; mt < 2; ++mt) {
            v8f acc = {0.f,0.f,0.f,0.f,0.f,0.f,0.f,0.f};
            const int m = mt * 16 + row;
            for (int k0 = 0; k0 < CCH; k0 += 4) {
                const int kb = k0 + khalf;
                v2f a, b;
                a[0] = pooled[m * CCH + kb]     * (1.0f / (float)HWN);
                a[1] = pooled[m * CCH + kb + 1] * (1.0f / (float)HWN);
                float bx = 0.f, by = 0.f;
                if (row < HIDDEN) { bx = w1[kb * HIDDEN + row]; by = w1[(kb + 1) * HIDDEN + row]; }
                b[0] = bx; b[1] = by;
                acc = __builtin_amdgcn_wmma_f32_16x16x4_f32(false, a, false, b,
                                                            (short)0, acc, false, false);
            }
            if (row < HIDDEN) {
                const float bias = b1v[row];
                #pragma unroll
                for (int r = 0; r < 8; ++r) {
                    float v = acc[r] + bias;
                    hbuf[(mt * 16 + half8 + r) * HIDDEN + row] = fmaxf(v, 0.0f);
                }
            }
        }
    }
    __syncthreads();

    // GEMM2: each wave handles one 16-wide N tile (16 waves x 16 = 256)
    {
        const int n0 = wid * 16;
        const int nn = n0 + row;
        for (int mt = 0; mt < 2; ++mt) {
            v8f acc = {0.f,0.f,0.f,0.f,0.f,0.f,0.f,0.f};
            const int m = mt * 16 + row;
            #pragma unroll
            for (int k0 = 0; k0 < HIDDEN; k0 += 4) {
                const int kb = k0 + khalf;
                v2f a, b;
                a[0] = hbuf[m * HIDDEN + kb];
                a[1] = hbuf[m * HIDDEN + kb + 1];
                b[0] = w2[kb * TWOKC + nn];
                b[1] = w2[(kb + 1) * TWOKC + nn];
                acc = __builtin_amdgcn_wmma_f32_16x16x4_f32(false, a, false, b,
                                                            (short)0, acc, false, false);
            }
            const float bias = b2v[nn];
            const int comp = nn & 1;
            const int c    = (nn >> 1) & 63;
            const int kbr  = nn >> 7;                 // K branch 0/1
            const float initv = (kbr == 0) ? 1.0f : 0.0f;
            const float lam   = (comp == 0) ? 1.0f : 0.5f;   // LAMBDA_ALPHA / LAMBDA_BETA
            #pragma unroll
            for (int r = 0; r < 8; ++r) {
                float v  = acc[r] + bias;
                float th = 2.0f / (1.0f + __expf(-v)) - 1.0f;
                int msamp = mt * 16 + half8 + r;
                coef[(msamp * CCH + c) * 4 + (kbr * 2 + comp)] = initv + lam * th;
            }
        }
    }
}

// ---------------- pass 2: out = max_k spatial*(x*a_k + b_k) ----------------
__global__ __launch_bounds__(256) void k_pass2(const float* __restrict__ x,
                                               const float* __restrict__ spatial,
                                               const float* __restrict__ coef,
                                               float* __restrict__ out) {
    const int chunk = blockIdx.x;          // 0..12
    const int n     = blockIdx.y;
    const int p     = chunk * 256 + threadIdx.x;
    if (p >= HW4) return;

    const float4* xn = (const float4*)x  + (size_t)n * CCH * HW4;
    float4*       on = (float4*)out      + (size_t)n * CCH * HW4;
    const float4  sp = ((const float4*)spatial)[(size_t)n * HW4 + p];
    const float4* cf = (const float4*)coef + n * CCH;

    for (int c = 0; c < CCH; ++c) {
        float4 cc = cf[c];                 // (a1, b1, a2, b2) — uniform, scalarized
        float4 v  = xn[(size_t)c * HW4 + p];
        float4 o;
        o.x = fmaxf(sp.x * (v.x * cc.x + cc.y), sp.x * (v.x * cc.z + cc.w));
        o.y = fmaxf(sp.y * (v.y * cc.x + cc.y), sp.y * (v.y * cc.z + cc.w));
        o.z = fmaxf(sp.z * (v.z * cc.x + cc.y), sp.z * (v.z * cc.z + cc.w));
        o.w = fmaxf(sp.w * (v.w * cc.x + cc.y), sp.w * (v.w * cc.z + cc.w));
        on[(size_t)c * HW4 + p] = o;
    }
}

extern "C" void kernel_launch(void* const* d_in, const int* in_sizes, int n_in,
                              void* d_out, int out_size, void* d_ws, size_t ws_size,
                              hipStream_t stream) {
    const float* x      = (const float*)d_in[0];
    const float* conv_w = (const float*)d_in[1];
    const float* conv_b = (const float*)d_in[2];
    const float* w1     = (const float*)d_in[3];
    const float* b1v    = (const float*)d_in[4];
    const float* w2     = (const float*)d_in[5];
    const float* b2v    = (const float*)d_in[6];
    float* out = (float*)d_out;

    float* ws     = (float*)d_ws;
    float* s_buf  = ws;                        // N*HWN floats (s, then spatial in place)
    float* pooled = ws + (size_t)NSAMP * HWN;  // N*C floats
    float* coef   = pooled + NSAMP * CCH;      // N*C*4 floats (16B aligned offset)

    k_zero<<<(NSAMP * CCH + 255) / 256, 256, 0, stream>>>(pooled);
    k_pass1<<<dim3(8, NSAMP), 256, 0, stream>>>(x, conv_w, conv_b, s_buf, pooled);
    k_softmax<<<NSAMP, 1024, 0, stream>>>(s_buf);
    k_coef<<<1, 512, 0, stream>>>(pooled, w1, b1v, w2, b2v, coef);
    k_pass2<<<dim3(13, NSAMP), 256, 0, stream>>>(x, s_buf, coef, out);
}